// ParallelMLPWithPE_52501680226675
// MI455X (gfx1250) — compile-verified
//
#include <hip/hip_runtime.h>
#include <math.h>

#define NUM_MLPS 50000
#define NSAMP    262144
#define PE_DIM   14
#define H1       64
#define H2       16
#define OUTD     3

typedef float v2f __attribute__((ext_vector_type(2)));
typedef float v8f __attribute__((ext_vector_type(8)));

__device__ __forceinline__ v8f wmma4(v2f a, v2f b, v8f c) {
  // V_WMMA_F32_16X16X4_F32: D(16x16,f32) = A(16x4,f32) * B(4x16,f32) + C
  return __builtin_amdgcn_wmma_f32_16x16x4_f32(
      /*neg_a=*/false, a, /*neg_b=*/false, b,
      /*c_mod=*/(short)0, c, /*reuse_a=*/false, /*reuse_b=*/false);
}

// ---------------- sorting infrastructure ----------------

__global__ void k_zero(int* __restrict__ p, int n) {
  int i = blockIdx.x * blockDim.x + threadIdx.x;
  if (i < n) p[i] = 0;
}

__global__ void k_hist(const int* __restrict__ idx, int* __restrict__ counts) {
  int i = blockIdx.x * blockDim.x + threadIdx.x;
  if (i < NSAMP) atomicAdd(&counts[idx[i]], 1);
}

// single-block exclusive scan over NUM_MLPS counters
__global__ void k_scan(const int* __restrict__ counts, int* __restrict__ offsets) {
  __shared__ int sd[1024];
  __shared__ int carry;
  int tid = threadIdx.x;
  if (tid == 0) carry = 0;
  __syncthreads();
  for (int base = 0; base < NUM_MLPS; base += 1024) {
    int i = base + tid;
    int v = (i < NUM_MLPS) ? counts[i] : 0;
    sd[tid] = v;
    __syncthreads();
    for (int off = 1; off < 1024; off <<= 1) {
      int t = (tid >= off) ? sd[tid - off] : 0;
      __syncthreads();
      sd[tid] += t;
      __syncthreads();
    }
    int incl = sd[tid];
    int c = carry;
    if (i < NUM_MLPS) offsets[i] = c + incl - v;
    __syncthreads();
    if (tid == 0) carry = c + sd[1023];
    __syncthreads();
  }
}

__global__ void k_scatter(const int* __restrict__ idx, const int* __restrict__ offsets,
                          int* __restrict__ cursor, int* __restrict__ sorted) {
  int i = blockIdx.x * blockDim.x + threadIdx.x;
  if (i < NSAMP) {
    int m = idx[i];
    int p = offsets[m] + atomicAdd(&cursor[m], 1);
    sorted[p] = i;
  }
}

// ---------------- main WMMA compute: one wave per MLP ----------------

__global__ void __launch_bounds__(32)
k_mlp(const float* __restrict__ x,
      const float* __restrict__ fc1w, const float* __restrict__ fc1b,
      const float* __restrict__ fc2w, const float* __restrict__ fc2b,
      const float* __restrict__ fc3w, const float* __restrict__ fc3b,
      const int* __restrict__ counts, const int* __restrict__ offsets,
      const int* __restrict__ sorted, float* __restrict__ out) {
  int mlp = blockIdx.x;
  int cnt = counts[mlp];
  if (cnt == 0) return;
  int start = offsets[mlp];

  int lane = threadIdx.x;
  int half = lane >> 4;   // 0: lanes 0-15, 1: lanes 16-31
  int lr   = lane & 15;   // row (A/D M-half select) / column (B,D N)

  const float* W1 = fc1w + (size_t)mlp * (PE_DIM * H1);
  const float* W2 = fc2w + (size_t)mlp * (H1 * H2);
  const float* W3 = fc3w + (size_t)mlp * (H2 * OUTD);

  // ---- preload B fragments (weights), loaded once per MLP ----
  // B(4x16) layout: lane = N (=lr); vgpr0/1 = K rows {0,1} (lanes 0-15) or {2,3} (lanes 16-31)
  v2f b1[4][4];
#pragma unroll
  for (int kt = 0; kt < 4; ++kt) {
#pragma unroll
    for (int ct = 0; ct < 4; ++ct) {
      int k0 = kt * 4 + 2 * half;
      int col = ct * 16 + lr;
      b1[kt][ct].x = (k0     < PE_DIM) ? W1[k0 * H1 + col]       : 0.0f;
      b1[kt][ct].y = (k0 + 1 < PE_DIM) ? W1[(k0 + 1) * H1 + col] : 0.0f;
    }
  }
  v2f b2[16];
#pragma unroll
  for (int kt = 0; kt < 16; ++kt) {
    int k0 = kt * 4 + 2 * half;
    b2[kt].x = W2[k0 * H2 + lr];
    b2[kt].y = W2[(k0 + 1) * H2 + lr];
  }
  v2f b3[4];
#pragma unroll
  for (int kt = 0; kt < 4; ++kt) {
    int k0 = kt * 4 + 2 * half;
    b3[kt].x = (lr < OUTD) ? W3[k0 * OUTD + lr]       : 0.0f;
    b3[kt].y = (lr < OUTD) ? W3[(k0 + 1) * OUTD + lr] : 0.0f;
  }
  // biases (C/D layout: lane = column -> broadcast across all 8 row-regs)
  float bias1[4];
#pragma unroll
  for (int ct = 0; ct < 4; ++ct) bias1[ct] = fc1b[(size_t)mlp * H1 + ct * 16 + lr];
  float bias2 = fc2b[(size_t)mlp * H2 + lr];
  float bias3 = (lr < OUTD) ? fc3b[(size_t)mlp * OUTD + lr] : 0.0f;

  __shared__ float h1s[16][H1 + 4];  // layer-1 activations, [sample][64]
  __shared__ float h2s[16][H2 + 4];  // layer-2 activations, [sample][16]

  for (int rb = 0; rb < cnt; rb += 16) {
    int rows = (cnt - rb < 16) ? (cnt - rb) : 16;

    // my A-row sample (dummy = first sample for padded rows; results not stored)
    int sidx = sorted[start + ((lr < rows) ? (rb + lr) : 0)];
    float x0 = x[2 * (size_t)sidx + 0];
    float x1 = x[2 * (size_t)sidx + 1];

    // positional encoding: [x, sin(f0 x), cos(f0 x), sin(f1 x), cos(f1 x), sin(f2 x), cos(f2 x)]
    // args are tiny (|f*x| <= 4*pi) -> hardware v_sin/v_cos path is plenty accurate
    float pe[16];
    pe[0] = x0; pe[1] = x1;
#pragma unroll
    for (int k = 0; k < 3; ++k) {
      float f = (float)(1 << k) * 3.14159265358979323846f;
      pe[2 + 4 * k + 0] = __sinf(f * x0);
      pe[2 + 4 * k + 1] = __sinf(f * x1);
      pe[2 + 4 * k + 2] = __cosf(f * x0);
      pe[2 + 4 * k + 3] = __cosf(f * x1);
    }
    pe[14] = 0.0f; pe[15] = 0.0f;

    // ---- layer 1: [16 x 14] @ [14 x 64] ----
#pragma unroll
    for (int ct = 0; ct < 4; ++ct) {
      float bb = bias1[ct];
      v8f acc = {bb, bb, bb, bb, bb, bb, bb, bb};
#pragma unroll
      for (int kt = 0; kt < 4; ++kt) {
        v2f a;
        a.x = half ? pe[kt * 4 + 2] : pe[kt * 4 + 0];
        a.y = half ? pe[kt * 4 + 3] : pe[kt * 4 + 1];
        acc = wmma4(a, b1[kt][ct], acc);
      }
      // D layout: lane = N (=lr), vgpr v -> M = v + 8*half
#pragma unroll
      for (int v = 0; v < 8; ++v) h1s[v + 8 * half][ct * 16 + lr] = acc[v];
    }
    __syncthreads();

    // ---- layer 2: [16 x 64] @ [64 x 16] ----
    v8f acc2 = {bias2, bias2, bias2, bias2, bias2, bias2, bias2, bias2};
#pragma unroll
    for (int kt = 0; kt < 16; ++kt) {
      v2f a;
      a.x = h1s[lr][kt * 4 + 2 * half + 0];
      a.y = h1s[lr][kt * 4 + 2 * half + 1];
      acc2 = wmma4(a, b2[kt], acc2);
    }
#pragma unroll
    for (int v = 0; v < 8; ++v) h2s[v + 8 * half][lr] = acc2[v];
    __syncthreads();

    // ---- layer 3: [16 x 16] @ [16 x 3(pad 16)] ----
    v8f acc3 = {bias3, bias3, bias3, bias3, bias3, bias3, bias3, bias3};
#pragma unroll
    for (int kt = 0; kt < 4; ++kt) {
      v2f a;
      a.x = h2s[lr][kt * 4 + 2 * half + 0];
      a.y = h2s[lr][kt * 4 + 2 * half + 1];
      acc3 = wmma4(a, b3[kt], acc3);
    }

    // sigmoid + scatter out; lane column lr in [0,3), row = v + 8*half
    if (lr < OUTD) {
#pragma unroll
      for (int v = 0; v < 8; ++v) {
        int r = v + 8 * half;
        if (r < rows) {
          int s = sorted[start + rb + r];
          float z = acc3[v];
          out[(size_t)s * OUTD + lr] = 1.0f / (1.0f + __expf(-z));
        }
      }
    }
    __syncthreads();  // protect h1s/h2s before next row-group
  }
}

// ---------------- launch ----------------

extern "C" void kernel_launch(void* const* d_in, const int* in_sizes, int n_in,
                              void* d_out, int out_size, void* d_ws, size_t ws_size,
                              hipStream_t stream) {
  (void)in_sizes; (void)n_in; (void)out_size; (void)ws_size;
  const float* x    = (const float*)d_in[0];
  const int*   idx  = (const int*)d_in[1];
  const float* fc1w = (const float*)d_in[2];
  const float* fc1b = (const float*)d_in[3];
  const float* fc2w = (const float*)d_in[4];
  const float* fc2b = (const float*)d_in[5];
  const float* fc3w = (const float*)d_in[6];
  const float* fc3b = (const float*)d_in[7];
  float* out = (float*)d_out;

  int* counts  = (int*)d_ws;
  int* offsets = counts + NUM_MLPS;
  int* cursor  = offsets + NUM_MLPS;
  int* sorted  = cursor + NUM_MLPS;

  k_zero<<<(3 * NUM_MLPS + 255) / 256, 256, 0, stream>>>(counts, 3 * NUM_MLPS);
  k_hist<<<(NSAMP + 255) / 256, 256, 0, stream>>>(idx, counts);
  k_scan<<<1, 1024, 0, stream>>>(counts, offsets);
  k_scatter<<<(NSAMP + 255) / 256, 256, 0, stream>>>(idx, offsets, cursor, sorted);
  k_mlp<<<NUM_MLPS, 32, 0, stream>>>(x, fc1w, fc1b, fc2w, fc2b, fc3w, fc3b,
                                     counts, offsets, sorted, out);
}